// EGNN_55276229099672
// MI455X (gfx1250) — compile-verified
//
#include <hip/hip_runtime.h>
#include <hip/hip_bf16.h>

// ---------------- problem constants ----------------
#define NN   25000
#define EE   400000
#define FF   128          // node feature dim (F = HID = 128 both layers)
#define DD   16           // edge attr dim
#define HH   64           // edge-MLP hidden
#define HIDC 128          // node hidden
#define OUTC 32
#define GGR  64           // graphs
#define K1   273          // 2*128 + 16 + 1
#define K1P  276          // padded to multiple of 4
#define CATW 284          // LDS row stride (284*4 % 8 == 0, gcd(28,64)=4 -> conflict-free)
#define K2   192          // node-MLP concat dim
#define CAT2W 196
#define HSTR 68           // h-tile stride

typedef float v2f __attribute__((ext_vector_type(2)));
typedef float v8f __attribute__((ext_vector_type(8)));

__device__ __forceinline__ float silu_f(float v) {
    return v * (1.0f / (1.0f + __expf(-v)));
}

// D = A(16x4,f32) * B(4x16,f32) + C(16x16,f32), wave32
__device__ __forceinline__ v8f wmma_f32(v2f a, v2f b, v8f c) {
    return __builtin_amdgcn_wmma_f32_16x16x4_f32(false, a, false, b, (short)0, c, false, false);
}

// =====================================================================
// Weight repack: (K,N) row-major -> K-pair-interleaved (P, N, 2) with
// zero padding for k >= K.  B fragment becomes one aligned b64 load.
// =====================================================================
__global__ __launch_bounds__(256) void repack_kernel(
    const float* __restrict__ src, float* __restrict__ dst,
    int K, int N, int P)
{
    int idx = blockIdx.x * 256 + threadIdx.x;
    if (idx >= P * N) return;
    int p = idx / N, n = idx - p * N;
    int k0 = 2 * p, k1 = 2 * p + 1;
    dst[idx * 2 + 0] = (k0 < K) ? src[k0 * N + n] : 0.f;
    dst[idx * 2 + 1] = (k1 < K) ? src[k1 * N + n] : 0.f;
}

// =====================================================================
// Edge kernel: 1 wave / block, 2 edge-tiles (32 edges) per wave.
// =====================================================================
__global__ __launch_bounds__(32) void egnn_edge_kernel(
    const float* __restrict__ x, const float* __restrict__ pos,
    const float* __restrict__ eattr, const long long* __restrict__ eidx,
    const float* __restrict__ wpM,  // packed (138,64,2), zero-padded K
    const float* __restrict__ wpE,  // packed (32,64,2)
    const float* __restrict__ ebv,  // (64,)
    const float* __restrict__ cw,   // (64,)
    const float* __restrict__ cb,   // (1,)
    float* __restrict__ e_agg, float* __restrict__ cu, float* __restrict__ deg)
{
    __shared__ float sCat[32][CATW];
    __shared__ float sH[32][HSTR];
    __shared__ float sDif[32][3];
    __shared__ int   sDst[32];

    const int lane = threadIdx.x;
    const int tileBase = blockIdx.x * 32;
    const int m    = lane & 15;
    const int gsel = lane >> 4;
    const int kk   = gsel * 2;

    // ---- gather cat rows ----
    for (int r = 0; r < 32; ++r) {
        int e = tileBase + r;                 // EE % 32 == 0
        int si = (int)eidx[e];
        int di = (int)eidx[EE + e];
        if (lane == 0) sDst[r] = di;
        if (lane < 3)  sDif[r][lane] = pos[di * 3 + lane] - pos[si * 3 + lane];
        for (int c = lane; c < FF; c += 32) {
            sCat[r][c]      = x[di * FF + c];
            sCat[r][FF + c] = x[si * FF + c];
        }
        if (lane < DD) sCat[r][2 * FF + lane] = eattr[e * DD + lane];
    }
    __syncthreads();
    {   // radial + zero pad (row = lane)
        float dx = sDif[lane][0], dy = sDif[lane][1], dz = sDif[lane][2];
        sCat[lane][272] = dx * dx + dy * dy + dz * dz;
        sCat[lane][273] = 0.f; sCat[lane][274] = 0.f; sCat[lane][275] = 0.f;
    }
    __syncthreads();

    // ---- GEMM1: (32x276) @ (276x64), unguarded (weights zero-padded) ----
    v8f acc[2][4] = {};
    for (int k = 0; k < K1P; k += 4) {
        int ka = k + kk;
        v2f a0 = *(const v2f*)&sCat[m][ka];
        v2f a1 = *(const v2f*)&sCat[16 + m][ka];
        const float* bp = wpM + (size_t)(ka >> 1) * (HH * 2);
        #pragma unroll
        for (int nt = 0; nt < 4; ++nt) {
            v2f b = *(const v2f*)&bp[(m + nt * 16) * 2];
            acc[0][nt] = wmma_f32(a0, b, acc[0][nt]);
            acc[1][nt] = wmma_f32(a1, b, acc[1][nt]);
        }
    }
    #pragma unroll
    for (int mt = 0; mt < 2; ++mt)
        #pragma unroll
        for (int nt = 0; nt < 4; ++nt)
            #pragma unroll
            for (int v = 0; v < 8; ++v)
                sH[mt * 16 + v + 8 * gsel][m + nt * 16] = silu_f(acc[mt][nt][v]);
    __syncthreads();

    // ---- GEMM2: (32x64) @ (64x64) ----
    v8f ea[2][4] = {};
    for (int k = 0; k < HH; k += 4) {
        int ka = k + kk;
        v2f a0 = *(const v2f*)&sH[m][ka];
        v2f a1 = *(const v2f*)&sH[16 + m][ka];
        const float* bp = wpE + (size_t)(ka >> 1) * (HH * 2);
        #pragma unroll
        for (int nt = 0; nt < 4; ++nt) {
            v2f b = *(const v2f*)&bp[(m + nt * 16) * 2];
            ea[0][nt] = wmma_f32(a0, b, ea[0][nt]);
            ea[1][nt] = wmma_f32(a1, b, ea[1][nt]);
        }
    }
    // bias + silu -> e_ij tile (reuse sCat memory, stride HSTR)
    float* et = &sCat[0][0];
    #pragma unroll
    for (int mt = 0; mt < 2; ++mt)
        #pragma unroll
        for (int nt = 0; nt < 4; ++nt) {
            float bn = ebv[m + nt * 16];
            #pragma unroll
            for (int v = 0; v < 8; ++v)
                et[(mt * 16 + v + 8 * gsel) * HSTR + (m + nt * 16)] = silu_f(ea[mt][nt][v] + bn);
        }
    __syncthreads();

    // ---- coord scalar + scatter atomics ----
    float cwa = cw[lane], cwb = cw[lane + 32];
    float cb0 = cb[0];
    for (int r = 0; r < 32; ++r) {
        float ev0 = et[r * HSTR + lane];
        float ev1 = et[r * HSTR + lane + 32];
        float dot = ev0 * cwa + ev1 * cwb;
        dot += __shfl_xor(dot, 16);
        dot += __shfl_xor(dot, 8);
        dot += __shfl_xor(dot, 4);
        dot += __shfl_xor(dot, 2);
        dot += __shfl_xor(dot, 1);
        float s = silu_f(dot + cb0);
        int di = sDst[r];
        atomicAdd(&e_agg[di * HH + lane],      ev0);
        atomicAdd(&e_agg[di * HH + lane + 32], ev1);
        if (lane < 3)  atomicAdd(&cu[di * 3 + lane], sDif[r][lane] * s);
        if (lane == 3) atomicAdd(&deg[di], 1.0f);
    }
}

// =====================================================================
// Node kernel: 1 wave / block, 2 node-tiles (32 nodes) per wave.
//   x_out = silu([x, e_agg] @ nw1 + b1) @ nw2 + b2
//   pos_out = pos_in + cu / max(deg,1)
// =====================================================================
__global__ __launch_bounds__(32) void egnn_node_kernel(
    const float* __restrict__ x_in, const float* __restrict__ pos_in,
    const float* __restrict__ e_agg, const float* __restrict__ cu,
    const float* __restrict__ deg,
    const float* __restrict__ wp1,  // packed (96,64,2)
    const float* __restrict__ b1,   // (64,)
    const float* __restrict__ wp2,  // packed (32,128,2)
    const float* __restrict__ b2,   // (128,)
    float* __restrict__ x_out, float* __restrict__ pos_out)
{
    __shared__ float sCat[32][CAT2W];
    __shared__ float sH[32][HSTR];

    const int lane = threadIdx.x;
    const int tileBase = blockIdx.x * 32;
    const int m    = lane & 15;
    const int gsel = lane >> 4;
    const int kk   = gsel * 2;

    // gather [x, e_agg] (clamp rows; stores guarded later)
    for (int r = 0; r < 32; ++r) {
        int n = tileBase + r; if (n >= NN) n = NN - 1;
        for (int c = lane; c < FF; c += 32) sCat[r][c]      = x_in[n * FF + c];
        for (int c = lane; c < HH; c += 32) sCat[r][FF + c] = e_agg[n * HH + c];
    }
    __syncthreads();

    // GEMM1: (32x192) @ (192x64)
    v8f acc[2][4] = {};
    for (int k = 0; k < K2; k += 4) {
        int ka = k + kk;
        v2f a0 = *(const v2f*)&sCat[m][ka];
        v2f a1 = *(const v2f*)&sCat[16 + m][ka];
        const float* bp = wp1 + (size_t)(ka >> 1) * (HH * 2);
        #pragma unroll
        for (int nt = 0; nt < 4; ++nt) {
            v2f b = *(const v2f*)&bp[(m + nt * 16) * 2];
            acc[0][nt] = wmma_f32(a0, b, acc[0][nt]);
            acc[1][nt] = wmma_f32(a1, b, acc[1][nt]);
        }
    }
    #pragma unroll
    for (int mt = 0; mt < 2; ++mt)
        #pragma unroll
        for (int nt = 0; nt < 4; ++nt) {
            float bn = b1[m + nt * 16];
            #pragma unroll
            for (int v = 0; v < 8; ++v)
                sH[mt * 16 + v + 8 * gsel][m + nt * 16] = silu_f(acc[mt][nt][v] + bn);
        }
    __syncthreads();

    // GEMM2: (32x64) @ (64x128)
    v8f oc[2][8] = {};
    for (int k = 0; k < HH; k += 4) {
        int ka = k + kk;
        v2f a0 = *(const v2f*)&sH[m][ka];
        v2f a1 = *(const v2f*)&sH[16 + m][ka];
        const float* bp = wp2 + (size_t)(ka >> 1) * (HIDC * 2);
        #pragma unroll
        for (int nt = 0; nt < 8; ++nt) {
            v2f b = *(const v2f*)&bp[(m + nt * 16) * 2];
            oc[0][nt] = wmma_f32(a0, b, oc[0][nt]);
            oc[1][nt] = wmma_f32(a1, b, oc[1][nt]);
        }
    }
    // store x_out with bias
    #pragma unroll
    for (int mt = 0; mt < 2; ++mt)
        #pragma unroll
        for (int nt = 0; nt < 8; ++nt) {
            int n = m + nt * 16;
            float bn = b2[n];
            #pragma unroll
            for (int v = 0; v < 8; ++v) {
                int row = tileBase + mt * 16 + v + 8 * gsel;
                if (row < NN) x_out[row * HIDC + n] = oc[mt][nt][v] + bn;
            }
        }
    // pos mean update (row = lane)
    {
        int rn = tileBase + lane;
        if (rn < NN) {
            float dg = fmaxf(deg[rn], 1.0f);
            #pragma unroll
            for (int c = 0; c < 3; ++c)
                pos_out[rn * 3 + c] = pos_in[rn * 3 + c] + cu[rn * 3 + c] / dg;
        }
    }
}

// =====================================================================
// Graph mean-pool accumulation
// =====================================================================
__global__ __launch_bounds__(128) void pool_kernel(
    const float* __restrict__ x2, const long long* __restrict__ batch,
    float* __restrict__ gsum, float* __restrict__ gcnt)
{
    int n = blockIdx.x;
    int b = (int)batch[n];
    int c = threadIdx.x;
    atomicAdd(&gsum[b * HIDC + c], x2[n * HIDC + c]);
    if (c == 0) atomicAdd(&gcnt[b], 1.0f);
}

// head: hid = relu(relu(gsum/cnt) @ w1 + b1)
__global__ __launch_bounds__(256) void headA_kernel(
    const float* __restrict__ gsum, const float* __restrict__ gcnt,
    const float* __restrict__ w1, const float* __restrict__ b1,
    float* __restrict__ hid)
{
    int idx = blockIdx.x * 256 + threadIdx.x;
    if (idx >= GGR * HIDC) return;
    int r = idx >> 7, c = idx & 127;
    float inv = 1.0f / fmaxf(gcnt[r], 1.0f);
    float s = 0.f;
    for (int k = 0; k < HIDC; ++k)
        s += fmaxf(gsum[r * HIDC + k] * inv, 0.f) * w1[k * HIDC + c];
    hid[idx] = fmaxf(s + b1[c], 0.f);
}

// out = hid @ w2 + b2
__global__ __launch_bounds__(256) void headB_kernel(
    const float* __restrict__ hid, const float* __restrict__ w2,
    const float* __restrict__ b2, float* __restrict__ out)
{
    int idx = blockIdx.x * 256 + threadIdx.x;
    if (idx >= GGR * OUTC) return;
    int r = idx >> 5, c = idx & 31;
    float s = 0.f;
    for (int k = 0; k < HIDC; ++k)
        s += hid[r * HIDC + k] * w2[k * OUTC + c];
    out[idx] = s + b2[c];
}

// =====================================================================
extern "C" void kernel_launch(void* const* d_in, const int* in_sizes, int n_in,
                              void* d_out, int out_size, void* d_ws, size_t ws_size,
                              hipStream_t stream) {
    (void)in_sizes; (void)n_in; (void)out_size; (void)ws_size;

    const float*     x     = (const float*)d_in[0];
    const float*     pos   = (const float*)d_in[1];
    const float*     eattr = (const float*)d_in[2];
    const long long* eidx  = (const long long*)d_in[3];
    const long long* batch = (const long long*)d_in[4];

    // per-layer params: 5 + l*9 : mlp_w, edge_w, edge_b, coord_w, coord_b,
    //                              node_w1, node_b1, node_w2, node_b2
    const float* P[2][9];
    for (int l = 0; l < 2; ++l)
        for (int i = 0; i < 9; ++i)
            P[l][i] = (const float*)d_in[5 + l * 9 + i];
    const float* ow1 = (const float*)d_in[23];
    const float* ob1 = (const float*)d_in[24];
    const float* ow2 = (const float*)d_in[25];
    const float* ob2 = (const float*)d_in[26];

    // workspace layout (floats)
    float* ws    = (float*)d_ws;
    float* e_agg = ws;                         // N*64
    float* cu    = e_agg + (size_t)NN * HH;    // N*3
    float* deg   = cu    + (size_t)NN * 3;     // N
    float* x1    = deg   + (size_t)NN;         // N*128
    float* pos1  = x1    + (size_t)NN * HIDC;  // N*3
    float* x2    = pos1  + (size_t)NN * 3;     // N*128
    float* pos2  = x2    + (size_t)NN * HIDC;  // N*3
    float* gsum  = pos2  + (size_t)NN * 3;     // G*128
    float* gcnt  = gsum  + (size_t)GGR * HIDC; // G
    float* hid   = gcnt  + (size_t)GGR;        // G*128

    // packed weights (pairs * N * 2)
    const int PM = K1P / 2, PE = HH / 2, PN1 = K2 / 2, PN2 = HH / 2;
    float* wp[2][4];
    float* wcur = hid + (size_t)GGR * HIDC;
    for (int l = 0; l < 2; ++l) {
        wp[l][0] = wcur; wcur += (size_t)PM  * HH * 2;    // mlp_w
        wp[l][1] = wcur; wcur += (size_t)PE  * HH * 2;    // edge_w
        wp[l][2] = wcur; wcur += (size_t)PN1 * HH * 2;    // node_w1
        wp[l][3] = wcur; wcur += (size_t)PN2 * HIDC * 2;  // node_w2
    }

    // ---- repack weights (runs in-graph, tiny) ----
    for (int l = 0; l < 2; ++l) {
        repack_kernel<<<dim3((PM  * HH   + 255) / 256), dim3(256), 0, stream>>>(P[l][0], wp[l][0], K1, HH,   PM);
        repack_kernel<<<dim3((PE  * HH   + 255) / 256), dim3(256), 0, stream>>>(P[l][1], wp[l][1], HH, HH,   PE);
        repack_kernel<<<dim3((PN1 * HH   + 255) / 256), dim3(256), 0, stream>>>(P[l][5], wp[l][2], K2, HH,   PN1);
        repack_kernel<<<dim3((PN2 * HIDC + 255) / 256), dim3(256), 0, stream>>>(P[l][7], wp[l][3], HH, HIDC, PN2);
    }

    const size_t zBytes  = ((size_t)NN * HH + (size_t)NN * 3 + NN) * sizeof(float);
    const size_t zBytesG = ((size_t)GGR * HIDC + GGR) * sizeof(float);

    dim3 eGrid(EE / 32), nGrid((NN + 31) / 32), blk(32);

    // ---- layer 0 ----
    hipMemsetAsync(e_agg, 0, zBytes, stream);
    egnn_edge_kernel<<<eGrid, blk, 0, stream>>>(x, pos, eattr, eidx,
        wp[0][0], wp[0][1], P[0][2], P[0][3], P[0][4], e_agg, cu, deg);
    egnn_node_kernel<<<nGrid, blk, 0, stream>>>(x, pos, e_agg, cu, deg,
        wp[0][2], P[0][6], wp[0][3], P[0][8], x1, pos1);

    // ---- layer 1 ----
    hipMemsetAsync(e_agg, 0, zBytes, stream);
    egnn_edge_kernel<<<eGrid, blk, 0, stream>>>(x1, pos1, eattr, eidx,
        wp[1][0], wp[1][1], P[1][2], P[1][3], P[1][4], e_agg, cu, deg);
    egnn_node_kernel<<<nGrid, blk, 0, stream>>>(x1, pos1, e_agg, cu, deg,
        wp[1][2], P[1][6], wp[1][3], P[1][8], x2, pos2);

    // ---- pool + head ----
    hipMemsetAsync(gsum, 0, zBytesG, stream);
    pool_kernel<<<dim3(NN), dim3(128), 0, stream>>>(x2, batch, gsum, gcnt);
    headA_kernel<<<dim3((GGR * HIDC + 255) / 256), dim3(256), 0, stream>>>(gsum, gcnt, ow1, ob1, hid);
    headB_kernel<<<dim3((GGR * OUTC + 255) / 256), dim3(256), 0, stream>>>(hid, ow2, ob2, (float*)d_out);
}